// CrossGAT_28836410425811
// MI455X (gfx1250) — compile-verified
//
#include <hip/hip_runtime.h>
#include <hip/hip_bf16.h>

// CrossGAT on MI455X (gfx1250).
// Dense lin layers via V_WMMA_F32_16X16X4_F32 (fp32-exact). W staged transposed
// in (dynamic) LDS with padded stride so both A and B fragments come from
// conflict-free ds_load_2addr_b64 ops; inner loop is pure DS + WMMA.
// Edge softmax/aggregation via L2-resident float atomics (all tensors << 192MB L2).
// h2@Wd_x collapsed to GEMV h2@(Wd_x@ad_x).

#define N1C 50000
#define N2C 50000
#define EC  800000
#define DC  128
#define LC  2
#define WSTR 132  // padded LDS row stride (floats): 132%64==4 -> conflict-free

typedef float v2f __attribute__((ext_vector_type(2)));
typedef float v8f __attribute__((ext_vector_type(8)));

__device__ __forceinline__ void atomicMaxF(float* addr, float val) {
  // ordered-int-bits float max (buffer initialized to -FLT_MAX)
  if (val >= 0.f) atomicMax((int*)addr, __float_as_int(val));
  else            atomicMin((unsigned int*)addr, (unsigned int)__float_as_int(val));
}

// ---------------------------------------------------------------------------
// C[N x 128] = A[N x 128] @ W[128 x 128] (+ bias). N must be multiple of 16.
// Block = 256 threads = 8 waves; block covers 16 rows, each wave one 16-col tile.
// Dynamic LDS: sWt[128 cols][WSTR] (transposed W) + sA[16 rows][WSTR].
// WMMA f32 16x16x4 fragment layout:
//   A: lane(0-15)=M, VGPR{0,1}=K{0,1}; lanes 16-31 same M, K{2,3}
//   B: lane(0-15)=N, VGPR{0,1}=K{0,1}; lanes 16-31 same N, K{2,3}
//   C: VGPR r -> row r (lanes 0-15) / r+8 (lanes 16-31), col = lane%16
// ---------------------------------------------------------------------------
__global__ __launch_bounds__(256)
void gemm_n128_wmma(const float* __restrict__ A, const float* __restrict__ W,
                    const float* __restrict__ bias, float* __restrict__ C) {
  extern __shared__ float smem[];
  float* sWt = smem;               // 128 * WSTR floats, sWt[n*WSTR + k] = W[k][n]
  float* sA  = smem + DC * WSTR;   // 16 * WSTR floats,  sA[r*WSTR + k] = A[row r][k]

  const int rowbase = blockIdx.x * 16;
  const int tid = threadIdx.x;

  // stage W transposed: 4096 float4 reads spread over 256 threads (16 each)
#pragma unroll
  for (int i = 0; i < 16; ++i) {
    const int c  = tid + 256 * i;      // float4 chunk id
    const int k  = c >> 5;             // W row
    const int n4 = (c & 31) * 4;       // W col base
    const float4 w4 = *(const float4*)(W + (size_t)k * DC + n4);
    sWt[(n4 + 0) * WSTR + k] = w4.x;
    sWt[(n4 + 1) * WSTR + k] = w4.y;
    sWt[(n4 + 2) * WSTR + k] = w4.z;
    sWt[(n4 + 3) * WSTR + k] = w4.w;
  }
  // stage 16x128 A tile: 512 float4, 2 per thread (padded rows are 16B aligned)
#pragma unroll
  for (int i = 0; i < 2; ++i) {
    const int c  = tid + 256 * i;
    const int r  = c >> 5;
    const int n4 = (c & 31) * 4;
    *(float4*)(sA + r * WSTR + n4) =
        *(const float4*)(A + (size_t)(rowbase + r) * DC + n4);
  }
  __syncthreads();

  const int lane = tid & 31;
  const int cb   = (tid >> 5) * 16;   // wave's column tile base
  const int m    = lane & 15;
  const int half = lane >> 4;

  const float* ap = sA + m * WSTR + half * 2;          // A[m][k], A[m][k+1]
  const float* bp = sWt + (cb + m) * WSTR + half * 2;  // W[k][cb+m], W[k+1][..]

  v8f acc = {};
#pragma unroll 8
  for (int kk = 0; kk < 32; ++kk) {
    const v2f a = *(const v2f*)(ap + kk * 4);  // ds_load (paired into 2addr_b64)
    const v2f b = *(const v2f*)(bp + kk * 4);
    acc = __builtin_amdgcn_wmma_f32_16x16x4_f32(false, a, false, b,
                                                (short)0, acc, false, false);
  }

  const int col = cb + m;
  const float bv = bias ? bias[col] : 0.f;
  // lane holds rows {r + 8*half}, col = cb+m; no bounds checks (N % 16 == 0)
  float* cp = C + ((size_t)rowbase + half * 8) * DC + col;
#pragma unroll
  for (int r = 0; r < 8; ++r) {
    cp[(size_t)r * DC] = acc[r] + bv;
  }
}

// o1[i] = dot(X[i,:], v1); optionally o2[i] = dot(X[i,:], v2). One wave/row.
__global__ __launch_bounds__(256)
void rowdot2_kernel(const float* __restrict__ X, const float* __restrict__ v1,
                    const float* __restrict__ v2, float* __restrict__ o1,
                    float* __restrict__ o2, int N) {
  const int row  = blockIdx.x * (blockDim.x >> 5) + (threadIdx.x >> 5);
  const int lane = threadIdx.x & 31;
  if (row >= N) return;
  const float4 x = ((const float4*)(X + (size_t)row * DC))[lane];
  const float4 a = ((const float4*)v1)[lane];
  float s1 = x.x * a.x + x.y * a.y + x.z * a.z + x.w * a.w;
  float s2 = 0.f;
  if (v2) {
    const float4 b = ((const float4*)v2)[lane];
    s2 = x.x * b.x + x.y * b.y + x.z * b.z + x.w * b.w;
  }
#pragma unroll
  for (int off = 16; off > 0; off >>= 1) {
    s1 += __shfl_down(s1, off, 32);
    s2 += __shfl_down(s2, off, 32);
  }
  if (lane == 0) {
    o1[row] = s1;
    if (o2) o2[row] = s2;
  }
}

// v[k] = dot(W[k,:], a)  (128x128 @ 128) — single small block
__global__ void vecmat128_kernel(const float* __restrict__ W,
                                 const float* __restrict__ a,
                                 float* __restrict__ v) {
  const int k = threadIdx.x;
  float s = 0.f;
#pragma unroll 8
  for (int j = 0; j < DC; ++j) s += W[(size_t)k * DC + j] * a[j];
  v[k] = s;
}

__global__ void fillf_kernel(float* p, float val, int n) {
  const int i = blockIdx.x * blockDim.x + threadIdx.x;
  if (i < n) p[i] = val;
}

// out[i, c] = b1[c] (+ b2[c])
__global__ void init_bias_kernel(float* __restrict__ out,
                                 const float* __restrict__ b1,
                                 const float* __restrict__ b2, int total) {
  const int i = blockIdx.x * blockDim.x + threadIdx.x;
  if (i >= total) return;
  const int c = i & (DC - 1);
  float v = b1[c];
  if (b2) v += b2[c];
  out[i] = v;
}

// pass A: logit = leaky_relu(as[src]+ad[dst], 0.2); segment max via atomics
__global__ void edge_logit_max_kernel(const float* __restrict__ as,
                                      const float* __restrict__ ad,
                                      const int* __restrict__ src,
                                      const int* __restrict__ dst,
                                      float* __restrict__ elog,
                                      float* __restrict__ maxb, int E) {
  const int e = blockIdx.x * blockDim.x + threadIdx.x;
  if (e >= E) return;
  float l = as[src[e]] + ad[dst[e]];
  l = l > 0.f ? l : 0.2f * l;
  elog[e] = l;
  atomicMaxF(maxb + dst[e], l);
}

// pass B: w = exp(logit - max[dst]); segment sum via atomics
__global__ void edge_exp_sum_kernel(const float* __restrict__ elog,
                                    const float* __restrict__ maxb,
                                    const int* __restrict__ dst,
                                    float* __restrict__ ew,
                                    float* __restrict__ sumb, int E) {
  const int e = blockIdx.x * blockDim.x + threadIdx.x;
  if (e >= E) return;
  const int d = dst[e];
  const float w = __expf(elog[e] - maxb[d]);
  ew[e] = w;
  atomicAdd(sumb + d, w);
}

// pass C: O[dst,:] += (w/sum[dst]) * G[src,:]  — one wave per edge, float4 gather
__global__ __launch_bounds__(256)
void edge_aggr_kernel(const float* __restrict__ ew, const float* __restrict__ sumb,
                      const int* __restrict__ src, const int* __restrict__ dst,
                      const float* __restrict__ G, float* __restrict__ O, int E) {
  const long gid = (long)blockIdx.x * blockDim.x + threadIdx.x;
  const int e    = (int)(gid >> 5);
  const int lane = (int)(gid & 31);
  if (e >= E) return;
  const int s = src[e], d = dst[e];
  const float coef = ew[e] / (sumb[d] + 1e-16f);
  const float4 g = *(const float4*)(G + (size_t)s * DC + lane * 4);
  float* op = O + (size_t)d * DC + lane * 4;
  atomicAdd(op + 0, coef * g.x);
  atomicAdd(op + 1, coef * g.y);
  atomicAdd(op + 2, coef * g.z);
  atomicAdd(op + 3, coef * g.w);
}

__global__ void relu_inplace_kernel(float* p, int n) {
  const int i = blockIdx.x * blockDim.x + threadIdx.x;
  if (i < n) {
    const float v = p[i];
    p[i] = v > 0.f ? v : 0.f;
  }
}

// ---------------------------------------------------------------------------
extern "C" void kernel_launch(void* const* d_in, const int* in_sizes, int n_in,
                              void* d_out, int out_size, void* d_ws, size_t ws_size,
                              hipStream_t stream) {
  const float* x1    = (const float*)d_in[0];
  const float* x2    = (const float*)d_in[1];
  const int*   ei11  = (const int*)d_in[2];
  const int*   ei22  = (const int*)d_in[3];
  const int*   ei12  = (const int*)d_in[4];
  const float* W_i1  = (const float*)d_in[5];
  const float* as_i1 = (const float*)d_in[6];
  const float* ad_i1 = (const float*)d_in[7];
  const float* b_i1  = (const float*)d_in[8];
  const float* W_i2  = (const float*)d_in[9];
  const float* as_i2 = (const float*)d_in[10];
  const float* ad_i2 = (const float*)d_in[11];
  const float* b_i2  = (const float*)d_in[12];
  const float* Ws_x  = (const float*)d_in[13];
  const float* Wd_x  = (const float*)d_in[14];
  const float* as_x  = (const float*)d_in[15];
  const float* ad_x  = (const float*)d_in[16];
  const float* b_x   = (const float*)d_in[17];
  const float* W_out = (const float*)d_in[18];
  const float* b_out = (const float*)d_in[19];

  float* ws = (float*)d_ws;
  size_t off = 0;
  auto alloc = [&](size_t n) { float* p = ws + off; off += n; return p; };

  float* h1   = alloc((size_t)N1C * DC);   // also o1 (aliased lifetime)
  float* h2   = alloc((size_t)N2C * DC);   // also o2
  float* g1   = alloc((size_t)N1C * DC);
  float* g2   = alloc((size_t)N2C * DC);
  float* gx   = alloc((size_t)N1C * DC);
  float* a_s1 = alloc(N1C);
  float* a_d1 = alloc(N1C);
  float* a_s2 = alloc(N2C);
  float* a_d2 = alloc(N2C);
  float* a_sx = alloc(N1C);
  float* a_dx = alloc(N2C);
  float* vd   = alloc(DC);
  float* maxb = alloc(N1C > N2C ? N1C : N2C);
  float* sumb = alloc(N1C > N2C ? N1C : N2C);
  float* elog = alloc(EC);
  float* ew   = alloc(EC);
  (void)ws_size; (void)in_sizes; (void)n_in; (void)out_size;

  const dim3 blk(256);
  const dim3 gemm_g1(N1C / 16), gemm_g2(N2C / 16);
  const dim3 rd_g1((N1C + 7) / 8), rd_g2((N2C + 7) / 8);
  const dim3 nd_g1((N1C * DC + 255) / 256), nd_g2((N2C * DC + 255) / 256);
  const dim3 seg_g1((N1C + 255) / 256), seg_g2((N2C + 255) / 256);
  const dim3 e_g((EC + 255) / 256);
  const dim3 ea_g(EC / 8);  // one wave per edge
  const size_t lds_bytes = (size_t)(DC * WSTR + 16 * WSTR) * sizeof(float);

  const float NEG_BIG = -3.402823466e38f;

  const float* in1 = x1;
  const float* in2 = x2;

  for (int l = 0; l < LC; ++l) {
    const size_t wo = (size_t)l * DC * DC;
    const size_t vo = (size_t)l * DC;

    // linear transforms (WMMA)
    gemm_n128_wmma<<<gemm_g1, blk, lds_bytes, stream>>>(in1, W_i1 + wo, nullptr, g1);
    gemm_n128_wmma<<<gemm_g2, blk, lds_bytes, stream>>>(in2, W_i2 + wo, nullptr, g2);
    gemm_n128_wmma<<<gemm_g1, blk, lds_bytes, stream>>>(in1, Ws_x + wo, nullptr, gx);

    // attention coefficients
    rowdot2_kernel<<<rd_g1, blk, 0, stream>>>(g1, as_i1 + vo, ad_i1 + vo, a_s1, a_d1, N1C);
    rowdot2_kernel<<<rd_g2, blk, 0, stream>>>(g2, as_i2 + vo, ad_i2 + vo, a_s2, a_d2, N2C);
    rowdot2_kernel<<<rd_g1, blk, 0, stream>>>(gx, as_x + vo, nullptr, a_sx, nullptr, N1C);
    vecmat128_kernel<<<1, DC, 0, stream>>>(Wd_x + wo, ad_x + vo, vd);
    rowdot2_kernel<<<rd_g2, blk, 0, stream>>>(in2, vd, nullptr, a_dx, nullptr, N2C);

    // output accumulators (alias h buffers; inputs to GEMMs already consumed)
    float* o1 = h1;
    float* o2 = h2;
    init_bias_kernel<<<nd_g1, blk, 0, stream>>>(o1, b_i1 + vo, nullptr, N1C * DC);
    init_bias_kernel<<<nd_g2, blk, 0, stream>>>(o2, b_i2 + vo, b_x + vo, N2C * DC);

    // --- edge set 1->1 (dst graph 1) ---
    fillf_kernel<<<seg_g1, blk, 0, stream>>>(maxb, NEG_BIG, N1C);
    fillf_kernel<<<seg_g1, blk, 0, stream>>>(sumb, 0.f, N1C);
    edge_logit_max_kernel<<<e_g, blk, 0, stream>>>(a_s1, a_d1, ei11, ei11 + EC, elog, maxb, EC);
    edge_exp_sum_kernel<<<e_g, blk, 0, stream>>>(elog, maxb, ei11 + EC, ew, sumb, EC);
    edge_aggr_kernel<<<ea_g, blk, 0, stream>>>(ew, sumb, ei11, ei11 + EC, g1, o1, EC);

    // --- edge set 2->2 (dst graph 2) ---
    fillf_kernel<<<seg_g2, blk, 0, stream>>>(maxb, NEG_BIG, N2C);
    fillf_kernel<<<seg_g2, blk, 0, stream>>>(sumb, 0.f, N2C);
    edge_logit_max_kernel<<<e_g, blk, 0, stream>>>(a_s2, a_d2, ei22, ei22 + EC, elog, maxb, EC);
    edge_exp_sum_kernel<<<e_g, blk, 0, stream>>>(elog, maxb, ei22 + EC, ew, sumb, EC);
    edge_aggr_kernel<<<ea_g, blk, 0, stream>>>(ew, sumb, ei22, ei22 + EC, g2, o2, EC);

    // --- edge set 1->2 (dst graph 2) ---
    fillf_kernel<<<seg_g2, blk, 0, stream>>>(maxb, NEG_BIG, N2C);
    fillf_kernel<<<seg_g2, blk, 0, stream>>>(sumb, 0.f, N2C);
    edge_logit_max_kernel<<<e_g, blk, 0, stream>>>(a_sx, a_dx, ei12, ei12 + EC, elog, maxb, EC);
    edge_exp_sum_kernel<<<e_g, blk, 0, stream>>>(elog, maxb, ei12 + EC, ew, sumb, EC);
    edge_aggr_kernel<<<ea_g, blk, 0, stream>>>(ew, sumb, ei12, ei12 + EC, gx, o2, EC);

    // activations (in place; o1/o2 ARE h1/h2)
    relu_inplace_kernel<<<nd_g1, blk, 0, stream>>>(o1, N1C * DC);
    relu_inplace_kernel<<<nd_g2, blk, 0, stream>>>(o2, N2C * DC);

    in1 = h1;
    in2 = h2;
  }

  // final projection, fused bias, straight into d_out (out1 | out2 concat)
  float* out = (float*)d_out;
  gemm_n128_wmma<<<gemm_g1, blk, lds_bytes, stream>>>(h1, W_out, b_out, out);
  gemm_n128_wmma<<<gemm_g2, blk, lds_bytes, stream>>>(h2, W_out, b_out,
                                                      out + (size_t)N1C * DC);
}